// SpiralConv_56221121905150
// MI455X (gfx1250) — compile-verified
//
#include <hip/hip_runtime.h>
#include <math.h>
#include <stdint.h>

// Problem dims
#define BB 4
#define LL 4096
#define DD 1024
#define CH 16              // scan chunk (= WMMA M/N)
#define TT 256             // t-tile per workgroup (= 16 chunks = one WMMA N-tile)
#define DT 32              // d-tile per workgroup (128B coalesced rows)
#define NT (LL / TT)       // 16 t-tiles
#define BD (BB * DD)       // 4096 sequences

typedef float v2f __attribute__((ext_vector_type(2)));
typedef float v8f __attribute__((ext_vector_type(8)));

#if defined(__has_builtin)
#if __has_builtin(__builtin_amdgcn_global_load_async_to_lds_b32) && \
    __has_builtin(__builtin_amdgcn_global_store_async_from_lds_b32) && \
    __has_builtin(__builtin_amdgcn_s_wait_asynccnt)
#define USE_ASYNC_LDS 1
#endif
#endif

typedef __attribute__((address_space(1))) int g_i32;   // global-side operand
typedef __attribute__((address_space(3))) int l_i32;   // LDS-side operand

__device__ __forceinline__ g_i32* to_global_as(const void* p) {
    // generic global pointers are bit-identical to addrspace(1) addresses
    return (g_i32*)(unsigned long long)(uintptr_t)p;
}
__device__ __forceinline__ l_i32* to_lds_as(const void* p) {
    // low 32 bits of a generic LDS pointer == LDS byte offset (ISA 10.2)
    return (l_i32*)(unsigned)(uintptr_t)p;
}

// ---------------------------------------------------------------------------
// K0: phazor powers. Planes [j*DD + d]: j = 0..16 -> phazor^j, j = 17 -> phazor^256.
// phazor = (p/|p|) * exp(-|p|)
// ---------------------------------------------------------------------------
__global__ void k_powers(const float* __restrict__ ph_re, const float* __restrict__ ph_im,
                         float* __restrict__ pow_re, float* __restrict__ pow_im) {
    int d = blockIdx.x * blockDim.x + threadIdx.x;
    if (d >= DD) return;
    float a = ph_re[d], b = ph_im[d];
    float mag = sqrtf(a * a + b * b);
    float s = __expf(-mag) / mag;
    float phr = a * s, phi = b * s;
    float cr = 1.f, ci = 0.f;
    float s16r = 1.f, s16i = 0.f;
    for (int j = 0; j <= 16; ++j) {
        pow_re[j * DD + d] = cr;
        pow_im[j * DD + d] = ci;
        if (j == 16) { s16r = cr; s16i = ci; }
        float nr = cr * phr - ci * phi;
        float ni = cr * phi + ci * phr;
        cr = nr; ci = ni;
    }
    // phazor^256 = (phazor^16)^(2^4)
    for (int k = 0; k < 4; ++k) {
        float nr = s16r * s16r - s16i * s16i;
        float ni = 2.f * s16r * s16i;
        s16r = nr; s16i = ni;
    }
    pow_re[17 * DD + d] = s16r;
    pow_im[17 * DD + d] = s16i;
}

// ---------------------------------------------------------------------------
// K1: per-(b,d,tile) zero-seeded driven recurrence over 256 steps ->
// tile-local end value. E = phazor*E + phazor_init*x. Coalesced over (b,d).
// ---------------------------------------------------------------------------
__global__ void k_tile_ends(const float* __restrict__ x,
                            const float* __restrict__ pii_re, const float* __restrict__ pii_im,
                            const float* __restrict__ pow_re, const float* __restrict__ pow_im,
                            float* __restrict__ tends_re, float* __restrict__ tends_im) {
    const int idx  = blockIdx.x * blockDim.x + threadIdx.x;  // 65536 threads
    const int bd   = idx & (BD - 1);
    const int tile = idx >> 12;                              // BD == 2^12
    const int d    = bd & (DD - 1);
    const int b    = bd >> 10;
    const float phr = pow_re[DD + d];     // phazor^1
    const float phi = pow_im[DD + d];
    const float pir = pii_re[d];
    const float pim = pii_im[d];
    float Er = 0.f, Ei = 0.f;
    size_t base = (size_t)(b * LL + tile * TT) * DD + d;
    for (int s = 0; s < TT; ++s) {
        const float xv = x[base + (size_t)s * DD];
        const float nr = phr * Er - phi * Ei + pir * xv;
        const float ni = phr * Ei + phi * Er + pim * xv;
        Er = nr; Ei = ni;
    }
    tends_re[(size_t)tile * BD + bd] = Er;
    tends_im[(size_t)tile * BD + bd] = Ei;
}

// ---------------------------------------------------------------------------
// K2: 16-step cross-tile scan per (b,d). Seed = last_conv, step = phazor^256.
// tcar(k) = carry entering tile k (= g at end of tile k-1).
// ---------------------------------------------------------------------------
__global__ void k_tile_scan(const float* __restrict__ lc_re, const float* __restrict__ lc_im,
                            const float* __restrict__ pow_re, const float* __restrict__ pow_im,
                            const float* __restrict__ tends_re, const float* __restrict__ tends_im,
                            float* __restrict__ tcar_re, float* __restrict__ tcar_im) {
    const int bd = blockIdx.x * blockDim.x + threadIdx.x;
    if (bd >= BD) return;
    const int d = bd & (DD - 1);
    const float qr = pow_re[17 * DD + d];  // phazor^256
    const float qi = pow_im[17 * DD + d];
    float Er = lc_re[bd], Ei = lc_im[bd];
    for (int k = 0; k < NT; ++k) {
        const size_t idx = (size_t)k * BD + bd;
        tcar_re[idx] = Er;
        tcar_im[idx] = Ei;
        const float nr = tends_re[idx] + qr * Er - qi * Ei;
        const float ni = tends_im[idx] + qr * Ei + qi * Er;
        Er = nr; Ei = ni;
    }
}

// ---------------------------------------------------------------------------
// K3: main WMMA kernel. grid=(NT, DD/DT, BB), 256 threads (8 waves x 4 dims).
// Per dim: local tiles via 8x V_WMMA_F32_16X16X4_F32; chunk ends = WMMA row 15;
// in-wave 16-step shuffle scan (q = phazor^16) seeded by tile carry; apply
// Re(phazor^{i+1} * carry) correction in registers; coalesced LDS-staged I/O
// via async global<->LDS (ASYNCcnt path).
// ---------------------------------------------------------------------------
__global__ void __launch_bounds__(256)
k_main(const float* __restrict__ x,
       const float* __restrict__ pii_re, const float* __restrict__ pii_im,
       const float* __restrict__ pow_re, const float* __restrict__ pow_im,
       const float* __restrict__ tcar_re, const float* __restrict__ tcar_im,
       float* __restrict__ out) {
    __shared__ float xs[TT][DT + 1];
    __shared__ float os[TT][DT + 1];

    const int tile = blockIdx.x;
    const int tt   = tile * TT;
    const int d0   = blockIdx.y * DT;
    const int b    = blockIdx.z;
    const int tid  = threadIdx.x;

    // ---- stage-in: async global->LDS (ASYNCcnt path) or plain fallback ----
    for (int i = tid; i < TT * DT; i += 256) {
        const int t = i >> 5, dl = i & 31;
        const size_t gi = (size_t)(b * LL + tt + t) * DD + d0 + dl;
#ifdef USE_ASYNC_LDS
        __builtin_amdgcn_global_load_async_to_lds_b32(
            to_global_as(x + gi), to_lds_as(&xs[t][dl]), 0, 0);
#else
        xs[t][dl] = x[gi];
#endif
    }
#ifdef USE_ASYNC_LDS
    __builtin_amdgcn_s_wait_asynccnt(0);
#endif
    __syncthreads();

    const int wave = tid >> 5;
    const int lane = tid & 31;
    const int mrow = lane & 15;   // A: M row / B,D: N column (= chunk)
    const int hi   = lane >> 4;   // K-pair / M+8 selector

    for (int j = 0; j < 4; ++j) {
        const int dl = wave * 4 + j;
        const int d  = d0 + dl;
        const float pir = pii_re[d];
        const float pim = pii_im[d];

        v8f acc_re = {};
        v8f acc_im = {};
        #pragma unroll
        for (int kb = 0; kb < 4; ++kb) {
            v2f a_re, a_im, bf;
            #pragma unroll
            for (int r = 0; r < 2; ++r) {
                const int K    = kb * 4 + hi * 2 + r;
                const int diff = mrow - K;
                float Pr = 0.f, Pi = 0.f;
                if (diff >= 0) {                      // lower-triangular
                    Pr = pow_re[diff * DD + d];
                    Pi = pow_im[diff * DD + d];
                }
                a_re[r] = pir * Pr - pim * Pi;        // Re(phazor_init*phazor^diff)
                a_im[r] = pim * Pr + pir * Pi;
                bf[r]   = xs[mrow * 16 + K][dl];      // X[k, chunk=mrow]
            }
            acc_re = __builtin_amdgcn_wmma_f32_16x16x4_f32(
                false, a_re, false, bf, (short)0, acc_re, false, false);
            acc_im = __builtin_amdgcn_wmma_f32_16x16x4_f32(
                false, a_im, false, bf, (short)0, acc_im, false, false);
        }

        // ---- in-wave chunk-carry scan. Chunk-local complex ends live in
        // acc_*[7] (M=15) of hi lanes (lane 16+c holds chunk c). All lanes
        // redundantly run the 16-step recurrence, capturing their own carry.
        const float q16r = pow_re[16 * DD + d];
        const float q16i = pow_im[16 * DD + d];
        const size_t tci = (size_t)tile * BD + (size_t)b * DD + d;
        float Er = tcar_re[tci];     // carry entering this 256-t tile
        float Ei = tcar_im[tci];
        float cRe = Er, cIm = Ei;    // carry for this lane's chunk (mrow)
        #pragma unroll
        for (int c = 0; c < 16; ++c) {
            if (c == mrow) { cRe = Er; cIm = Ei; }
            const float er = __shfl(acc_re[7], 16 + c, 32);
            const float ei = __shfl(acc_im[7], 16 + c, 32);
            const float nr = er + q16r * Er - q16i * Ei;
            const float ni = ei + q16r * Ei + q16i * Er;
            Er = nr; Ei = ni;
        }

        // ---- corrected output tile: out = local_re + Re(phazor^{M+1}*carry)
        #pragma unroll
        for (int v = 0; v < 8; ++v) {
            const int m = v + 8 * hi;
            const float Pr = pow_re[(m + 1) * DD + d];
            const float Pi = pow_im[(m + 1) * DD + d];
            os[mrow * 16 + m][dl] = acc_re[v] + Pr * cRe - Pi * cIm;
        }
    }
    __syncthreads();

    // ---- stage-out: async LDS->global or plain fallback ----
    for (int i = tid; i < TT * DT; i += 256) {
        const int t = i >> 5, dl = i & 31;
        const size_t gi = (size_t)(b * LL + tt + t) * DD + d0 + dl;
#ifdef USE_ASYNC_LDS
        __builtin_amdgcn_global_store_async_from_lds_b32(
            to_global_as(out + gi), to_lds_as(&os[t][dl]), 0, 0);
#else
        out[gi] = os[t][dl];
#endif
    }
#ifdef USE_ASYNC_LDS
    __builtin_amdgcn_s_wait_asynccnt(0);
#endif
}

// ---------------------------------------------------------------------------
extern "C" void kernel_launch(void* const* d_in, const int* in_sizes, int n_in,
                              void* d_out, int out_size, void* d_ws, size_t ws_size,
                              hipStream_t stream) {
    const float* x     = (const float*)d_in[0];
    const float* ph_re = (const float*)d_in[1];
    const float* ph_im = (const float*)d_in[2];
    const float* pi_re = (const float*)d_in[3];
    const float* pi_im = (const float*)d_in[4];
    const float* lc_re = (const float*)d_in[5];
    const float* lc_im = (const float*)d_in[6];
    float* out = (float*)d_out;

    // Workspace (floats): powers 2*18*DD, tile ends 2*NT*BD, tile carries 2*NT*BD
    // total ~1.2 MB
    float* ws       = (float*)d_ws;
    float* pow_re   = ws;
    float* pow_im   = pow_re + 18 * DD;
    float* tends_re = pow_im + 18 * DD;
    float* tends_im = tends_re + (size_t)NT * BD;
    float* tcar_re  = tends_im + (size_t)NT * BD;
    float* tcar_im  = tcar_re + (size_t)NT * BD;

    k_powers<<<dim3(DD / 256), dim3(256), 0, stream>>>(ph_re, ph_im, pow_re, pow_im);

    k_tile_ends<<<dim3((NT * BD) / 256), dim3(256), 0, stream>>>(
        x, pi_re, pi_im, pow_re, pow_im, tends_re, tends_im);

    k_tile_scan<<<dim3(BD / 256), dim3(256), 0, stream>>>(
        lc_re, lc_im, pow_re, pow_im, tends_re, tends_im, tcar_re, tcar_im);

    k_main<<<dim3(NT, DD / DT, BB), dim3(256), 0, stream>>>(
        x, pi_re, pi_im, pow_re, pow_im, tcar_re, tcar_im, out);
}